// loss_1271310319647
// MI455X (gfx1250) — compile-verified
//
#include <hip/hip_runtime.h>
#include <cstdint>

typedef __attribute__((ext_vector_type(2))) float v2f;
typedef __attribute__((ext_vector_type(8))) float v8f;

#define NBOX 64
#define BOXF 12              // dwords per box record (48B, b128-friendly total)
#define NBLK 256
#define NTHR 256
#define IMG_W 512
#define IMG_H 512
#define NPIX (IMG_W * IMG_H)

// ws layout (dwords):
//   [0, NBOX*BOXF)                : box records
//   [NBOX*BOXF, NBOX*BOXF+NBLK*2) : per-block sums of c+s (lanes 0-15, lanes 16-31 of each wave)

// ---------------------------------------------------------------------------
// Kernel 1: per-box precompute (K = 64 boxes)
// record: {x1f,y1f,x2f,y2f, ax=x1+cx, ay=y1+cy, -0.5/R, log(h), cx_bits, cy_bits, 0, 0}
// ---------------------------------------------------------------------------
__global__ void prep_boxes(const float* __restrict__ ann,
                           const int* __restrict__ stride_p,
                           float* __restrict__ ws) {
  int k = threadIdx.x;
  if (k >= NBOX) return;
  const float inv_s = 1.0f / (float)stride_p[0];
  const float a0 = ann[k * 4 + 0];
  const float a1 = ann[k * 4 + 1];
  const float a2 = ann[k * 4 + 2];
  const float a3 = ann[k * 4 + 3];
  const int x1 = (int)floorf(a0 * inv_s);
  const int y1 = (int)floorf(a1 * inv_s);
  const int x2 = (int)floorf(a2 * inv_s);
  const int y2 = (int)floorf(a3 * inv_s);
  const int cx = (x1 + x2) >> 1;
  const int cy = (y1 + y2) >> 1;
  const float cxf = (float)cx, cyf = (float)cy;
  const float R = sqrtf(cxf * cxf + cyf * cyf);
  const float hgt = (float)(y2 - y1);

  float* r = ws + k * BOXF;
  r[0] = (float)x1;
  r[1] = (float)y1;
  r[2] = (float)x2;
  r[3] = (float)y2;
  r[4] = (float)x1 + cxf;       // reference uses X - x1 - cx
  r[5] = (float)y1 + cyf;
  r[6] = -0.5f / R;
  r[7] = __logf(hgt);
  r[8] = __int_as_float(cx);
  r[9] = __int_as_float(cy);
  r[10] = 0.0f;
  r[11] = 0.0f;
}

// ---------------------------------------------------------------------------
// Kernel 2: per-pixel losses + WMMA wave reduction
// ---------------------------------------------------------------------------
__global__ void __launch_bounds__(NTHR) pixel_loss(
    const float* __restrict__ cmap, const float* __restrict__ smap,
    const float* __restrict__ ws, float* __restrict__ partials) {
  __shared__ float sbox[NBOX * BOXF];   // 3072 bytes

  const int tid = threadIdx.x;

  // Async copy of the box table into LDS (ASYNCcnt path): 192 lanes x 16B.
  if (tid < (NBOX * BOXF) / 4) {
    uint32_t lds_off = (uint32_t)(uintptr_t)(&sbox[0]) + (uint32_t)tid * 16u;
    uint32_t voff = (uint32_t)tid * 16u;
    uint64_t gbase = (uint64_t)(uintptr_t)ws;
    asm volatile("global_load_async_to_lds_b128 %0, %1, %2"
                 :: "v"(lds_off), "v"(voff), "s"(gbase)
                 : "memory");
  }
  asm volatile("s_wait_asynccnt 0" ::: "memory");
  __syncthreads();

  float acc_c = 0.0f, acc_s = 0.0f;

  // 262144 / (256*256) = 4 iterations, exactly uniform -> EXEC all-ones later.
  for (int i = blockIdx.x * NTHR + tid; i < NPIX; i += NBLK * NTHR) {
    const int xi = i & (IMG_W - 1);
    const int yi = i >> 9;
    const float X = (float)xi, Y = (float)yi;

    float p = cmap[i];
    p = fminf(fmaxf(p, 1.0e-4f), 1.0f - 1.0e-4f);
    const float sv = smap[i];

    float gauss = 0.0f;
    bool anybox = false;
    bool is_pos = false;
    int best_key = -1;
    float sgt = 0.0f;

    for (int k = 0; k < NBOX; ++k) {
      const float* r = &sbox[k * BOXF];   // same addr across lanes -> LDS broadcast
      const float x1 = r[0], y1 = r[1], x2 = r[2], y2 = r[3];
      const bool inb = (X >= x1) && (X < x2) && (Y >= y1) && (Y < y2);
      if (inb) {  // EXEC-mask skip: ~99% of (pixel,box) pairs avoid sqrt+exp
        anybox = true;
        const float dx = X - r[4];
        const float dy = Y - r[5];
        const float dist = sqrtf(dx * dx + dy * dy);
        const float g = __expf(r[6] * dist);
        gauss = fmaxf(gauss, g);
      }
      const int cx = __float_as_int(r[8]);
      const int cy = __float_as_int(r[9]);
      const int ox = xi - cx;
      const int oy = yi - cy;
      is_pos = is_pos || ((ox == 0) && (oy == 0));
      // scale_gt scatter: positions (cy+o, cx+o), o in [-2,2]; last write wins
      // in flattened (o,k) order -> keep lexicographic max (o,k).
      if ((ox == oy) && (ox >= -2) && (ox <= 2)) {
        const int key = (ox + 2) * NBOX + k;
        if (key > best_key) { best_key = key; sgt = r[7]; }
      }
    }

    // center (focal BCE) term
    const float q = 1.0f - gauss;
    const float q2 = q * q;
    const float gpow = q2 * q2 * (anybox ? 1.0f : 0.0f);  // (1-gauss)^BETA * pos
    const float alpha = is_pos ? 1.0f : gpow;             // ALPHA = 1.0
    const float pw = is_pos ? (1.0f - p) : p;             // GAMMA = 2.0
    const float bce = is_pos ? -__logf(p) : -__logf(1.0f - p);
    acc_c += alpha * pw * pw * bce;

    // scale (smooth L1) term, only where scale_gt != 0 (log h >= log 8 > 0)
    if (best_key >= 0) {
      const float d = fabsf(sgt - sv);
      acc_s += (d <= 1.0f) ? (0.5f * d * d) : (d - 0.5f);
    }
  }

  // -------- wave reduction via V_WMMA_F32_16X16X4_F32 (exact: f32, 0/1 B) ----
  // A (16x4 f32, 2 VGPRs/lane): A[m][0]=acc_c[m], A[m][1]=acc_s[m],
  //                             A[m][2]=acc_c[m+16], A[m][3]=acc_s[m+16]
  // B routes: col0 <- K0+K1 (c+s of lanes 0-15), col1 <- K2+K3 (c+s of lanes 16-31):
  //   B[0][0]=B[1][0]=1 (lane0: b0,b1), B[2][1]=B[3][1]=1 (lane17: b0,b1)
  const int lane = tid & 31;
  v2f A; A.x = acc_c; A.y = acc_s;
  const float bb = (lane == 0 || lane == 17) ? 1.0f : 0.0f;
  v2f Bm; Bm.x = bb; Bm.y = bb;
  v8f Cm = {0.f, 0.f, 0.f, 0.f, 0.f, 0.f, 0.f, 0.f};
  v8f Dm = __builtin_amdgcn_wmma_f32_16x16x4_f32(
      /*neg_a=*/false, A, /*neg_b=*/false, Bm,
      /*c_mod=*/(short)0, Cm, /*reuse_a=*/false, /*reuse_b=*/false);
  // Fold D over M: lane l holds D[m][l%16] for m in {0..7}+8*(l>=16).
  float t = Dm[0] + Dm[1] + Dm[2] + Dm[3] + Dm[4] + Dm[5] + Dm[6] + Dm[7];
  t += __shfl_xor(t, 16, 32);   // lane0: c+s summed over lanes 0-15; lane1: over lanes 16-31

  __shared__ float wsum[2][NTHR / 32];
  const int wv = tid >> 5;
  if (lane == 0) wsum[0][wv] = t;
  if (lane == 1) wsum[1][wv] = t;
  __syncthreads();
  if (tid == 0) {
    float c = 0.0f, s = 0.0f;
    for (int w = 0; w < NTHR / 32; ++w) { c += wsum[0][w]; s += wsum[1][w]; }
    partials[blockIdx.x * 2 + 0] = c;
    partials[blockIdx.x * 2 + 1] = s;
  }
}

// ---------------------------------------------------------------------------
// Kernel 3: deterministic final reduction (single wave32)
// ---------------------------------------------------------------------------
__global__ void finalize(const float* __restrict__ partials,
                         float* __restrict__ out) {
  const int lane = threadIdx.x;
  float c = 0.0f, s = 0.0f;
  for (int j = lane; j < NBLK; j += 32) {
    c += partials[j * 2 + 0];
    s += partials[j * 2 + 1];
  }
  for (int off = 16; off >= 1; off >>= 1) {
    c += __shfl_xor(c, off, 32);
    s += __shfl_xor(s, off, 32);
  }
  if (lane == 0) {
    out[0] = c * (1.0f / (float)NBOX);   // / max(1, K)
    out[1] = s * (1.0f / (float)NBOX);
  }
}

// ---------------------------------------------------------------------------
extern "C" void kernel_launch(void* const* d_in, const int* in_sizes, int n_in,
                              void* d_out, int out_size, void* d_ws,
                              size_t ws_size, hipStream_t stream) {
  const float* cmap = (const float*)d_in[0];  // (16,1,512,512), batch 0 used
  const float* smap = (const float*)d_in[1];  // (16,1,512,512), batch 0 used
  const float* ann  = (const float*)d_in[2];  // (16,64,4), batch 0 used
  const int*   strd = (const int*)d_in[3];    // scalar stride (=4)
  (void)in_sizes; (void)n_in; (void)out_size; (void)ws_size;

  float* ws = (float*)d_ws;
  float* partials = ws + NBOX * BOXF;

  prep_boxes<<<1, 64, 0, stream>>>(ann, strd, ws);
  pixel_loss<<<NBLK, NTHR, 0, stream>>>(cmap, smap, ws, partials);
  finalize<<<1, 32, 0, stream>>>(partials, (float*)d_out);
}